// PredictorGCN_25898652795267
// MI455X (gfx1250) — compile-verified
//
#include <hip/hip_runtime.h>
#include <hip/hip_bf16.h>
#include <math.h>

#define NNODES 200000
#define NEDGES 6400000
#define INDIM  128
#define DDIM   16
#define NGRAPH 1024
#define BN_EPS 1e-5f

typedef __attribute__((ext_vector_type(2))) float v2f;
typedef __attribute__((ext_vector_type(8))) float v8f;

// ---------------- utility fills ----------------
__global__ __launch_bounds__(256) void fill_kernel(float* __restrict__ p, int n, float v) {
    int i = blockIdx.x * 256 + threadIdx.x;
    if (i < n) p[i] = v;
}

// ---------------- degree + rsqrt ----------------
__global__ __launch_bounds__(256) void deg_kernel(const int* __restrict__ col,
                                                  float* __restrict__ deg, int E) {
    int e = blockIdx.x * 256 + threadIdx.x;
    if (e < E) atomicAdd(&deg[col[e]], 1.0f);
}

__global__ __launch_bounds__(256) void rsqrt_kernel(float* __restrict__ d, int n) {
    int i = blockIdx.x * 256 + threadIdx.x;
    if (i < n) d[i] = rsqrtf(d[i] + 1.0f);  // +1 self-loop
}

// ---------------- WMMA GEMM: H[nrows,16] = X[nrows,K] @ W[K,16] ----------------
// One wave per 16-row tile, full D=16 columns. fp32 path: V_WMMA_F32_16X16X4_F32.
// A 16x4 f32 layout (ISA 7.12.2): lanes 0-15 hold M=0..15, VGPR0/1 = K pair,
// lane-half selects K offset of 2.  B 4x16: K in VGPR dim, N=lane%16.
template <int K>
__global__ __launch_bounds__(256) void gemm_wmma_kernel(const float* __restrict__ X,
                                                        const float* __restrict__ W,
                                                        float* __restrict__ H,
                                                        int nrows) {
    int wave    = (blockIdx.x * 256 + threadIdx.x) >> 5;
    int lane    = threadIdx.x & 31;
    int rowBase = wave << 4;
    if (rowBase >= nrows) return;           // wave-uniform: EXEC stays all-ones for WMMA

    int half = lane >> 4;                   // 0 or 1
    int mn   = lane & 15;                   // M for A-frag, N for B/C-frag

    const float* xrow = X + (size_t)(rowBase + mn) * K + 2 * half;
    v8f c = {};
#pragma unroll
    for (int k = 0; k < K; k += 4) {
        v2f a = *(const v2f*)(xrow + k);    // (M=mn, K=k+2*half .. +1), 8B aligned
        v2f b;
        b.x = W[(k + 2 * half) * DDIM + mn];
        b.y = W[(k + 2 * half + 1) * DDIM + mn];
        c = __builtin_amdgcn_wmma_f32_16x16x4_f32(
                /*neg_a=*/false, a, /*neg_b=*/false, b,
                /*c_mod=*/(short)0, c, /*reuse_a=*/false, /*reuse_b=*/false);
    }
    // C/D 16x16 f32: VGPR v holds (M = rowBase + v + 8*half, N = mn)
#pragma unroll
    for (int v = 0; v < 8; ++v)
        H[(size_t)(rowBase + v + 8 * half) * DDIM + mn] = c[v];
}

// ---------------- edge aggregation (scatter-add, L2-resident atomics) ----------------
__global__ __launch_bounds__(256) void agg_kernel(const int* __restrict__ row,
                                                  const int* __restrict__ col,
                                                  const float* __restrict__ dis,
                                                  const float* __restrict__ H,
                                                  float* __restrict__ A, int E) {
    int idx = blockIdx.x * 256 + threadIdx.x;   // E*16 = 102.4M < 2^31
    if (idx >= E * DDIM) return;
    int e = idx >> 4;
    int f = idx & 15;
    int r = row[e], d = col[e];
    float coef = dis[r] * dis[d];
    atomicAdd(&A[(size_t)d * DDIM + f], H[(size_t)r * DDIM + f] * coef);
}

// ---------------- fused self-loop + bias + ReLU + BatchNorm (in place) ----------------
__global__ __launch_bounds__(256) void finalize_kernel(float* __restrict__ A,       // agg in / x out
                                                       const float* __restrict__ H, // xW
                                                       const float* __restrict__ dis,
                                                       const float* __restrict__ bias,
                                                       const float* __restrict__ gam,
                                                       const float* __restrict__ bet,
                                                       const float* __restrict__ mu,
                                                       const float* __restrict__ var,
                                                       int n) {
    int i = blockIdx.x * 256 + threadIdx.x;     // n = N*16 = 3.2M
    if (i >= n) return;
    int f = i & 15;
    int node = i >> 4;
    float d = dis[node];
    float val = A[i] + H[i] * (d * d) + bias[f];
    val = fmaxf(val, 0.0f);
    A[i] = (val - mu[f]) * rsqrtf(var[f] + BN_EPS) * gam[f] + bet[f];
}

// ---------------- pooling ----------------
__device__ inline void atomicMaxF(float* addr, float val) {
    // signed-int max for non-negative floats, unsigned min for negative floats
    if (val >= 0.0f) atomicMax((int*)addr, __float_as_int(val));
    else             atomicMin((unsigned int*)addr, __float_as_uint(val));
}

__global__ __launch_bounds__(256) void pool_kernel(const float* __restrict__ X2,
                                                   const int* __restrict__ batch,
                                                   float* __restrict__ gmax,
                                                   float* __restrict__ gsum,
                                                   float* __restrict__ cnt, int n) {
    int i = blockIdx.x * 256 + threadIdx.x;     // N*16
    if (i >= n) return;
    int node = i >> 4;
    int f = i & 15;
    int g = batch[node];
    float v = X2[i];
    atomicMaxF(&gmax[g * DDIM + f], v);
    atomicAdd(&gsum[g * DDIM + f], v);
    if (f == 0) atomicAdd(&cnt[g], 1.0f);
}

// ---------------- MLP head: [gmax|gmean] @ Wb -> relu -> @ Wm -> sigmoid ----------------
__global__ __launch_bounds__(256) void head_kernel(const float* __restrict__ gmax,
                                                   const float* __restrict__ gsum,
                                                   const float* __restrict__ cnt,
                                                   const float* __restrict__ Wb,
                                                   const float* __restrict__ bb,
                                                   const float* __restrict__ Wm,
                                                   const float* __restrict__ bm,
                                                   float* __restrict__ out, int G) {
    int g = blockIdx.x * 256 + threadIdx.x;
    if (g >= G) return;
    float z[2 * DDIM];
    float c = cnt[g];
#pragma unroll
    for (int f = 0; f < DDIM; ++f) {
        z[f]        = gmax[g * DDIM + f];
        z[DDIM + f] = gsum[g * DDIM + f] / c;
    }
    float acc = bm[0];
#pragma unroll
    for (int j = 0; j < 16; ++j) {
        float e = bb[j];
#pragma unroll
        for (int i = 0; i < 2 * DDIM; ++i) e += z[i] * Wb[i * 16 + j];
        acc += fmaxf(e, 0.0f) * Wm[j];
    }
    out[g] = 1.0f / (1.0f + expf(-acc));
}

// ---------------- launch ----------------
extern "C" void kernel_launch(void* const* d_in, const int* in_sizes, int n_in,
                              void* d_out, int out_size, void* d_ws, size_t ws_size,
                              hipStream_t stream) {
    const float* x    = (const float*)d_in[0];
    const int*   erow = (const int*)d_in[1];           // edge_index[0]
    const int*   ecol = ((const int*)d_in[1]) + NEDGES; // edge_index[1]
    const int*   batch = (const int*)d_in[2];
    const float* W1 = (const float*)d_in[3];
    const float* b1 = (const float*)d_in[4];
    const float* g1 = (const float*)d_in[5];
    const float* be1 = (const float*)d_in[6];
    const float* m1 = (const float*)d_in[7];
    const float* v1 = (const float*)d_in[8];
    const float* W2 = (const float*)d_in[9];
    const float* b2 = (const float*)d_in[10];
    const float* g2 = (const float*)d_in[11];
    const float* be2 = (const float*)d_in[12];
    const float* m2 = (const float*)d_in[13];
    const float* v2 = (const float*)d_in[14];
    const float* Wb = (const float*)d_in[15];
    const float* bb = (const float*)d_in[16];
    const float* Wm = (const float*)d_in[17];
    const float* bm = (const float*)d_in[18];
    float* out = (float*)d_out;

    // workspace layout (all 256B-aligned offsets)
    float* dis  = (float*)d_ws;                 // N
    float* bufA = dis + NNODES;                 // N*16 : h1 then h2
    float* bufB = bufA + (size_t)NNODES * DDIM; // N*16 : agg1/x1 then agg2/x2
    float* gmax = bufB + (size_t)NNODES * DDIM; // G*16
    float* gsum = gmax + NGRAPH * DDIM;         // G*16
    float* cnt  = gsum + NGRAPH * DDIM;         // G

    const int NF   = NNODES * DDIM;            // 3.2M
    const int EF   = NEDGES * DDIM;            // 102.4M
    const int blkN   = (NNODES + 255) / 256;
    const int blkE   = (NEDGES + 255) / 256;
    const int blkNF  = (NF + 255) / 256;
    const int blkEF  = (EF + 255) / 256;
    const int gemmBlocks = ((NNODES / 16) * 32 + 255) / 256;  // wave per 16-row tile

    // degrees (shared by both layers)
    fill_kernel<<<blkN, 256, 0, stream>>>(dis, NNODES, 0.0f);
    deg_kernel<<<blkE, 256, 0, stream>>>(ecol, dis, NEDGES);
    rsqrt_kernel<<<blkN, 256, 0, stream>>>(dis, NNODES);

    // ---- layer 1 ----
    gemm_wmma_kernel<INDIM><<<gemmBlocks, 256, 0, stream>>>(x, W1, bufA, NNODES);
    fill_kernel<<<blkNF, 256, 0, stream>>>(bufB, NF, 0.0f);
    agg_kernel<<<blkEF, 256, 0, stream>>>(erow, ecol, dis, bufA, bufB, NEDGES);
    finalize_kernel<<<blkNF, 256, 0, stream>>>(bufB, bufA, dis, b1, g1, be1, m1, v1, NF);

    // ---- layer 2 ----
    gemm_wmma_kernel<DDIM><<<gemmBlocks, 256, 0, stream>>>(bufB, W2, bufA, NNODES);
    fill_kernel<<<blkNF, 256, 0, stream>>>(bufB, NF, 0.0f);
    agg_kernel<<<blkEF, 256, 0, stream>>>(erow, ecol, dis, bufA, bufB, NEDGES);
    finalize_kernel<<<blkNF, 256, 0, stream>>>(bufB, bufA, dis, b2, g2, be2, m2, v2, NF);

    // ---- pooling + head ----
    fill_kernel<<<(NGRAPH * DDIM + 255) / 256, 256, 0, stream>>>(gmax, NGRAPH * DDIM, -INFINITY);
    fill_kernel<<<(NGRAPH * DDIM + 255) / 256, 256, 0, stream>>>(gsum, NGRAPH * DDIM, 0.0f);
    fill_kernel<<<(NGRAPH + 255) / 256, 256, 0, stream>>>(cnt, NGRAPH, 0.0f);
    pool_kernel<<<blkNF, 256, 0, stream>>>(bufB, batch, gmax, gsum, cnt, NF);
    head_kernel<<<(NGRAPH + 255) / 256, 256, 0, stream>>>(gmax, gsum, cnt, Wb, bb, Wm, bm, out, NGRAPH);
}